// HashEncodeFn_16587163697988
// MI455X (gfx1250) — compile-verified
//
#include <hip/hip_runtime.h>

// ---------------------------------------------------------------------------
// Multi-resolution hash-grid encode (Instant-NGP style), MI455X / gfx1250.
//  N = in_sizes[0]/3 points, 16 levels, 2 features/level -> out [N,32] f32.
//  Levels 0..2 dense (17^3, 33^3, 65^3), levels 3..15 hashed into 2^19 slots.
//
// Roofline: output stream 256 MB (NT stores, ~11us @ 23.3 TB/s) dominates HBM;
// 57 MB embedding table lives in the 192 MB L2, so the ~2 GB of random 8 B
// corner gathers are L2-resident. Level-0 table (39 KB) is DMA'd into LDS via
// the CDNA5 async global->LDS path (ASYNCcnt) once per block.
// ---------------------------------------------------------------------------

#ifndef __has_builtin
#define __has_builtin(x) 0
#endif

#if __has_builtin(__builtin_amdgcn_global_load_async_to_lds_b64)
#define ASYNC_LDS 1
#else
#define ASYNC_LDS 0
#endif

#define PRIME_Y   2654435761u
#define PRIME_Z   805459861u
#define HASH_MASK 0x7FFFFu          // 2^19 - 1

#define L0_N 4913u                  // 17^3 entries (39,304 B -> LDS)
#define OFF1 4913u                  // level-1 table offset (rows)
#define OFF2 40850u                 // level-2 table offset
#define OFF3 315475u                // first hashed level offset
#define HASH_N 524288u              // 2^19 rows per hashed level

typedef float __attribute__((ext_vector_type(4))) fvec4;
typedef int   __attribute__((ext_vector_type(2))) v2i;

// Address-space-qualified pointer types matching the async-LDS builtin:
//   arg0: v2i in addrspace(1) (global / "__device__")
//   arg1: v2i in addrspace(3) (LDS / "__shared__")
typedef __attribute__((address_space(1))) v2i* as1_v2i_p;
typedef __attribute__((address_space(3))) v2i* as3_v2i_p;

__device__ __forceinline__ void axis_setup(float x, float scale,
                                           unsigned& b, float& f) {
  float pos = x * scale + 0.5f;
  float pg  = floorf(pos);
  f = pos - pg;
  b = (unsigned)pg;                 // pos >= 0 for inputs in [0,1)
}

__device__ __forceinline__ void acc8(float2& a, float2 e, float w) {
  a.x = fmaf(w, e.x, a.x);
  a.y = fmaf(w, e.y, a.y);
}

// Dense level: idx = x + y*stride + z*stride^2 (always in-range, modulo-free).
__device__ __forceinline__ float2 level_dense(const float2* __restrict__ tab,
                                              float x, float y, float z,
                                              float scale, unsigned stride) {
  unsigned bx, by, bz; float fx, fy, fz;
  axis_setup(x, scale, bx, fx);
  axis_setup(y, scale, by, fy);
  axis_setup(z, scale, bz, fz);
  unsigned s2 = stride * stride;
  unsigned x0 = bx,          x1 = bx + 1u;
  unsigned y0 = by * stride, y1 = y0 + stride;
  unsigned z0 = bz * s2,     z1 = z0 + s2;
  float wx1 = fx, wx0 = 1.0f - fx;
  float wy1 = fy, wy0 = 1.0f - fy;
  float wz1 = fz, wz0 = 1.0f - fz;
  // Issue all 8 gathers up front for maximum memory-level parallelism.
  float2 e0 = tab[x0 + y0 + z0], e1 = tab[x1 + y0 + z0];
  float2 e2 = tab[x0 + y1 + z0], e3 = tab[x1 + y1 + z0];
  float2 e4 = tab[x0 + y0 + z1], e5 = tab[x1 + y0 + z1];
  float2 e6 = tab[x0 + y1 + z1], e7 = tab[x1 + y1 + z1];
  float2 a; a.x = 0.0f; a.y = 0.0f;
  acc8(a, e0, wx0 * wy0 * wz0);  acc8(a, e1, wx1 * wy0 * wz0);
  acc8(a, e2, wx0 * wy1 * wz0);  acc8(a, e3, wx1 * wy1 * wz0);
  acc8(a, e4, wx0 * wy0 * wz1);  acc8(a, e5, wx1 * wy0 * wz1);
  acc8(a, e6, wx0 * wy1 * wz1);  acc8(a, e7, wx1 * wy1 * wz1);
  return a;
}

// Hashed level: idx = (x ^ y*PRIME_Y ^ z*PRIME_Z) & (2^19-1).
__device__ __forceinline__ float2 level_hash(const float2* __restrict__ tab,
                                             float x, float y, float z,
                                             float scale) {
  unsigned bx, by, bz; float fx, fy, fz;
  axis_setup(x, scale, bx, fx);
  axis_setup(y, scale, by, fy);
  axis_setup(z, scale, bz, fz);
  unsigned x0  = bx,            x1  = bx + 1u;
  unsigned hy0 = by * PRIME_Y,  hy1 = hy0 + PRIME_Y;   // uint32 wrap == ref
  unsigned hz0 = bz * PRIME_Z,  hz1 = hz0 + PRIME_Z;
  float wx1 = fx, wx0 = 1.0f - fx;
  float wy1 = fy, wy0 = 1.0f - fy;
  float wz1 = fz, wz0 = 1.0f - fz;
  float2 e0 = tab[(x0 ^ hy0 ^ hz0) & HASH_MASK];
  float2 e1 = tab[(x1 ^ hy0 ^ hz0) & HASH_MASK];
  float2 e2 = tab[(x0 ^ hy1 ^ hz0) & HASH_MASK];
  float2 e3 = tab[(x1 ^ hy1 ^ hz0) & HASH_MASK];
  float2 e4 = tab[(x0 ^ hy0 ^ hz1) & HASH_MASK];
  float2 e5 = tab[(x1 ^ hy0 ^ hz1) & HASH_MASK];
  float2 e6 = tab[(x0 ^ hy1 ^ hz1) & HASH_MASK];
  float2 e7 = tab[(x1 ^ hy1 ^ hz1) & HASH_MASK];
  float2 a; a.x = 0.0f; a.y = 0.0f;
  acc8(a, e0, wx0 * wy0 * wz0);  acc8(a, e1, wx1 * wy0 * wz0);
  acc8(a, e2, wx0 * wy1 * wz0);  acc8(a, e3, wx1 * wy1 * wz0);
  acc8(a, e4, wx0 * wy0 * wz1);  acc8(a, e5, wx1 * wy0 * wz1);
  acc8(a, e6, wx0 * wy1 * wz1);  acc8(a, e7, wx1 * wy1 * wz1);
  return a;
}

__global__ __launch_bounds__(256)
void hashgrid_encode_kernel(const float* __restrict__ xyz,
                            const float* __restrict__ emb,
                            float* __restrict__ out, int N) {
  __shared__ float2 l0tab[L0_N];                       // 39,304 B
  const float2* etab = (const float2*)emb;

  // ---- Stage level-0 dense table into LDS via async DMA (ASYNCcnt path) ----
  for (unsigned i = threadIdx.x; i < L0_N; i += 256u) {
#if ASYNC_LDS
    __builtin_amdgcn_global_load_async_to_lds_b64(
        (as1_v2i_p)(etab + i),
        (as3_v2i_p)(&l0tab[i]),
        /*offset=*/0, /*cpol=*/0);
#else
    l0tab[i] = etab[i];
#endif
  }
#if ASYNC_LDS
#if __has_builtin(__builtin_amdgcn_s_wait_asynccnt)
  __builtin_amdgcn_s_wait_asynccnt(0);
#else
  asm volatile("s_wait_asynccnt 0" ::: "memory");
#endif
#endif
  __syncthreads();

  int n = (int)(blockIdx.x * 256u + threadIdx.x);
  if (n >= N) return;

  // Normalize [-bound,bound] -> [0,1]  (bound = 1)
  float px = (xyz[3 * n + 0] + 1.0f) * 0.5f;
  float py = (xyz[3 * n + 1] + 1.0f) * 0.5f;
  float pz = (xyz[3 * n + 2] + 1.0f) * 0.5f;

  float feat[32] __attribute__((aligned(16)));

  float2 r;
  r = level_dense(l0tab,       px, py, pz, 15.0f, 17u); feat[0] = r.x; feat[1] = r.y;
  r = level_dense(etab + OFF1, px, py, pz, 31.0f, 33u); feat[2] = r.x; feat[3] = r.y;
  r = level_dense(etab + OFF2, px, py, pz, 63.0f, 65u); feat[4] = r.x; feat[5] = r.y;

#pragma unroll
  for (int l = 3; l < 16; ++l) {
    float scale = (float)(16 << l) - 1.0f;              // exact: 2^l*16 - 1
    r = level_hash(etab + OFF3 + (unsigned)(l - 3) * HASH_N, px, py, pz, scale);
    feat[2 * l]     = r.x;
    feat[2 * l + 1] = r.y;
  }

  // 128 B per thread, each lane fully covers its own cacheline.
  // Non-temporal: keep the 256 MB output stream from evicting the 57 MB
  // embedding tables resident in the 192 MB L2.
  fvec4*       op = (fvec4*)(out + (size_t)n * 32);
  const fvec4* fp = (const fvec4*)feat;
#pragma unroll
  for (int i = 0; i < 8; ++i)
    __builtin_nontemporal_store(fp[i], op + i);
}

extern "C" void kernel_launch(void* const* d_in, const int* in_sizes, int n_in,
                              void* d_out, int out_size, void* d_ws, size_t ws_size,
                              hipStream_t stream) {
  const float* xyz = (const float*)d_in[0];   // [N,3] f32
  const float* emb = (const float*)d_in[1];   // [7131219,2] f32
  float* out = (float*)d_out;                 // [N,32] f32
  int N = in_sizes[0] / 3;
  int blocks = (N + 255) / 256;
  hipLaunchKernelGGL(hashgrid_encode_kernel, dim3(blocks), dim3(256), 0, stream,
                     xyz, emb, out, N);
}